// yoloLoss_40913858462360
// MI455X (gfx1250) — compile-verified
//
#include <hip/hip_runtime.h>

typedef __attribute__((ext_vector_type(2))) float v2f;
typedef __attribute__((ext_vector_type(8))) float v8f;

#define LAMBDA_LOC   10.0f
#define LAMBDA_NOOBJ 0.5f
#define CELL_C       26

__global__ void yolo_zero_kernel(float* out) {
    if (threadIdx.x == 0) out[0] = 0.0f;
}

__device__ __forceinline__ float iou_pair(float px, float py, float pw, float ph,
                                          float tx, float ty, float tw, float th) {
    float cx1 = px * 64.0f,  cy1 = py * 64.0f;
    float hw1 = pw * 224.0f, hh1 = ph * 224.0f;
    float cx2 = tx * 64.0f,  cy2 = ty * 64.0f;
    float hw2 = tw * 224.0f, hh2 = th * 224.0f;
    float l1 = cx1 - hw1, t1 = cy1 - hh1, r1 = cx1 + hw1, b1 = cy1 + hh1;
    float l2 = cx2 - hw2, t2 = cy2 - hh2, r2 = cx2 + hw2, b2 = cy2 + hh2;
    float xA = fmaxf(l1, l2), yA = fmaxf(t1, t2);
    float xB = fminf(r1, r2), yB = fminf(b1, b2);
    float inter = fmaxf(0.0f, xB - xA + 1.0f) * fmaxf(0.0f, yB - yA + 1.0f);
    float a1 = (r1 - l1 + 1.0f) * (b1 - t1 + 1.0f);
    float a2 = (r2 - l2 + 1.0f) * (b2 - t2 + 1.0f);
    return inter / (a1 + a2 - inter);
}

__global__ __launch_bounds__(256) void yolo_loss_kernel(const float* __restrict__ p,
                                                        const float* __restrict__ t,
                                                        float* __restrict__ out,
                                                        int nCells) {
    int idx = blockIdx.x * blockDim.x + threadIdx.x;
    float acc = 0.0f;

    if (idx < nCells) {
        // Row base: idx*26 floats = idx*104 bytes, always 8B-aligned -> float2 loads.
        const float2* pr = reinterpret_cast<const float2*>(p + (size_t)idx * CELL_C);
        const float2* tr = reinterpret_cast<const float2*>(t + (size_t)idx * CELL_C);
        float P[CELL_C], T[CELL_C];
        #pragma unroll
        for (int i = 0; i < CELL_C / 2; ++i) {
            float2 a = pr[i];
            float2 b = tr[i];
            P[2 * i] = a.x; P[2 * i + 1] = a.y;
            T[2 * i] = b.x; T[2 * i + 1] = b.y;
        }

        float iou1 = iou_pair(P[0], P[1], P[2], P[3], T[0], T[1], T[2], T[3]);
        float iou2 = iou_pair(P[5], P[6], P[7], P[8], T[0], T[1], T[2], T[3]);
        bool  best2   = iou2 > iou1;
        float iou_max = best2 ? iou2 : iou1;

        float vp0 = best2 ? P[5] : P[0];
        float vp1 = best2 ? P[6] : P[1];
        float vp2 = best2 ? P[7] : P[2];
        float vp3 = best2 ? P[8] : P[3];
        float vp4 = best2 ? P[9] : P[4];
        float vt0 = best2 ? T[5] : T[0];
        float vt1 = best2 ? T[6] : T[1];
        float vt2 = best2 ? T[7] : T[2];
        float vt3 = best2 ? T[8] : T[3];

        float obj   = T[4];
        float noobj = 1.0f - obj;

        float dx = vp0 - vt0, dy = vp1 - vt1;
        float xy = dx * dx + dy * dy;
        float dw = __builtin_sqrtf(vp2) - __builtin_sqrtf(vt2);
        float dh = __builtin_sqrtf(vp3) - __builtin_sqrtf(vt3);
        float wh = dw * dw + dh * dh;
        float doc = vp4 - iou_max;
        float oc  = doc * doc;
        float dn1 = P[4] - T[4];
        float dn2 = P[9] - T[9];
        float cls = 0.0f;
        #pragma unroll
        for (int c = 10; c < CELL_C; ++c) {
            float d = P[c] - T[c];
            cls += d * d;
        }

        acc = LAMBDA_LOC * obj * (xy + wh)
            + obj * oc
            + LAMBDA_NOOBJ * noobj * (dn1 * dn1 + dn2 * dn2)
            + 2.0f * obj * cls;
    }

    // ---- Wave-32 reduction via V_WMMA_F32_16X16X4_F32 (EXEC all-1s here) ----
    // A[m][k] layout: k=0 -> lane m reg0, k=1 -> lane m reg1,
    //                 k=2 -> lane m+16 reg0, k=3 -> lane m+16 reg1.
    // Set reg0 = acc, reg1 = 0, B = all ones:
    //   D[m][n] = acc(lane m) + acc(lane m+16)  for every n.
    v2f a;  a[0] = acc;  a[1] = 0.0f;
    v2f b;  b[0] = 1.0f; b[1] = 1.0f;
    v8f c = {0.0f, 0.0f, 0.0f, 0.0f, 0.0f, 0.0f, 0.0f, 0.0f};
    v8f d = __builtin_amdgcn_wmma_f32_16x16x4_f32(
        /*neg_a=*/false, a, /*neg_b=*/false, b,
        /*c_mod=*/(short)0, c, /*reuse_a=*/false, /*reuse_b=*/false);

    // Per C/D layout: lane L<16 holds rows M=0..7 (VGPR r -> M=r), lane L>=16 rows M=8..15.
    float s = d[0] + d[1] + d[2] + d[3] + d[4] + d[5] + d[6] + d[7];
    // s(lane<16) = sum_{m=0..7}(acc_m + acc_{m+16}); s(lane>=16) = sum_{m=8..15}(...).
    s += __shfl_xor(s, 16, 32);   // now every lane holds the full wave sum

    __shared__ float wsum[8];
    int lane = threadIdx.x & 31;
    int wid  = threadIdx.x >> 5;
    if (lane == 0) wsum[wid] = s;
    __syncthreads();
    if (threadIdx.x == 0) {
        float tsum = 0.0f;
        #pragma unroll
        for (int i = 0; i < 8; ++i) tsum += wsum[i];
        atomicAdd(out, tsum);
    }
}

extern "C" void kernel_launch(void* const* d_in, const int* in_sizes, int n_in,
                              void* d_out, int out_size, void* d_ws, size_t ws_size,
                              hipStream_t stream) {
    const float* predict = (const float*)d_in[0];
    const float* target  = (const float*)d_in[1];
    float* out = (float*)d_out;

    int nCells = in_sizes[0] / CELL_C;   // 16384 * 49 = 802816

    yolo_zero_kernel<<<1, 64, 0, stream>>>(out);

    int threads = 256;
    int blocks  = (nCells + threads - 1) / threads;
    yolo_loss_kernel<<<blocks, threads, 0, stream>>>(predict, target, out, nCells);
}